// FkMigrationLinear_33225867002346
// MI455X (gfx1250) — compile-verified
//
#include <hip/hip_runtime.h>

// ---------------------------------------------------------------------------
// FK (Stolt) migration for a linear array, CDNA5 / gfx1250.
// B=8, C=1, D=128, T=2048, F=1025, Y=X=256.
// All three dense stages (time-DFT, band einsum, image einsum) run on
// V_WMMA_F32_16X16X4_F32. Complex ops use re/im row-stacking (2 real WMMAs
// per complex K-step). Phase twiddles use f64 argument reduction + hardware
// v_sin/v_cos (branch-free, args < 3.3e3 rad).
// Workspace requirement: ~33.8 MB (tables aliased after use).
// ---------------------------------------------------------------------------

typedef float v2f __attribute__((ext_vector_type(2)));
typedef float v8f __attribute__((ext_vector_type(8)));

#define TWO_PI      6.28318530717958647692f
#define OMEGA_SCALE (TWO_PI / (2048.0f * 2.5e-8f))   // 2*pi*f_index -> rad/s
#define KX_SCALE    (TWO_PI / (128.0f * 3.0e-4f))    // fftfreq index -> rad/m
#define INV_C       (1.0f / 1540.0f)
#define NB    8
#define ND    128
#define NT    2048
#define NF    1025
#define NFP   1028     // K-padding for the band einsum (multiple of 4)
#define NFP2  1040     // N-padding for the time DFT (multiple of 16)
#define NROW  1024     // B*D rows
#define NY_   256
#define NX_   256

// Branch-free cos/sin: f64 reduction to revolutions, then HW sin/cos
// (llvm.amdgcn.sin/cos take the angle in revolutions and apply fract).
__device__ __forceinline__ void fast_cis(float theta, float* s, float* c) {
  double r = (double)theta * 1.5915494309189535e-1;   // /(2*pi)
  r -= floor(r);                                      // [0,1), exact fract
  const float fr = (float)r;
  *s = __builtin_amdgcn_sinf(fr);
  *c = __builtin_amdgcn_cosf(fr);
}

// -------------------------------------------------------------------- K0 ---
// Per-batch mean / inv-std over (D,T); 1/(apod_sum*fw_sum) at stats[16].
__global__ __launch_bounds__(256) void k_stats(const float* __restrict__ sino,
                                               float* __restrict__ stats) {
  __shared__ float ssum[256], ssq[256];
  const int b = blockIdx.x, tid = threadIdx.x;
  const float* p = sino + (size_t)b * (ND * NT);
  float s = 0.f, q = 0.f;
  for (int i = tid; i < ND * NT; i += 256) {
    float x = p[i];
    s += x;
    q = fmaf(x, x, q);
  }
  ssum[tid] = s; ssq[tid] = q;
  __syncthreads();
  for (int off = 128; off > 0; off >>= 1) {
    if (tid < off) { ssum[tid] += ssum[tid + off]; ssq[tid] += ssq[tid + off]; }
    __syncthreads();
  }
  if (tid == 0) {
    const float inv_n = 1.0f / (float)(ND * NT);
    float mean = ssum[0] * inv_n;
    float var  = ssq[0] * inv_n - mean * mean;
    stats[b]     = mean;
    stats[8 + b] = rsqrtf(var + 1.1920929e-7f);
    if (b == 0) {
      float ap = 0.f;
      for (int i = 0; i < ND; ++i)
        ap += 0.5f - 0.5f * __builtin_amdgcn_cosf((float)i * (1.0f / 127.0f));
      stats[16] = 1.0f / (fmaxf(ap, 1.1920929e-7f) * 2048.0f);  // fw_sum=2048
    }
  }
}

// ------------------------------------------------------------------- K0b ---
// Transposed pack: At[t][row] = (sino[row][t]-mean)*istd*hann(d). 32x32 tiles.
__global__ __launch_bounds__(256) void k_pack(const float* __restrict__ sino,
                                              const float* __restrict__ stats,
                                              float* __restrict__ At) {
  __shared__ float sm[32 * 33];
  const int rowbase = (blockIdx.x & 31) << 5;       // 1024/32 row tiles
  const int tbase   = (blockIdx.x >> 5) << 5;       // 2048/32 time tiles
  const int tx = threadIdx.x & 31, ty = threadIdx.x >> 5;  // ty in 0..7
#pragma unroll
  for (int i = 0; i < 4; ++i) {
    const int rl  = ty + i * 8;
    const int row = rowbase + rl;
    const int b = row >> 7, d = row & 127;
    const float mean = stats[b], istd = stats[8 + b];
    const float apod = 0.5f - 0.5f * __builtin_amdgcn_cosf((float)d * (1.0f / 127.0f));
    const float x = sino[(size_t)row * NT + tbase + tx];
    sm[tx * 33 + rl] = (x - mean) * istd * apod;    // sm[t_local][row_local]
  }
  __syncthreads();
#pragma unroll
  for (int i = 0; i < 4; ++i) {
    const int tl = ty + i * 8;
    At[(size_t)(tbase + tl) * NROW + rowbase + tx] = sm[tl * 33 + tx];
  }
}

// ------------------------------------------------------------------- K0c ---
// DFT twiddle tables: twc[t][f]=cos, tws[t][f]=-sin of 2*pi*(f*t mod 2048)/2048.
// f padded to 1040 with zeros (matmul N padding).
__global__ __launch_bounds__(256) void k_twgen(float* __restrict__ twc,
                                               float* __restrict__ tws) {
  const int t = blockIdx.x;
  for (int f = threadIdx.x; f < NFP2; f += 256) {
    float c = 0.f, ns = 0.f;
    if (f < NF) {
      const unsigned idx = ((unsigned)f * (unsigned)t) & (NT - 1);
      const float rev = (float)idx * (1.0f / 2048.0f);
      c  =  __builtin_amdgcn_cosf(rev);
      ns = -__builtin_amdgcn_sinf(rev);
    }
    twc[(size_t)t * NFP2 + f] = c;
    tws[(size_t)t * NFP2 + f] = ns;
  }
}

// -------------------------------------------------------------------- K1 ---
// Time DFT as f32 WMMA: spec[row][f] = sum_t At[t][row] * e^{-i 2pi f t/T}.
// Each wave: 4 M-tiles (64 rows) x 1 N-tile (16 f), K=2048 in steps of 4.
__global__ __launch_bounds__(256) void k_time_wmma(const float* __restrict__ At,
                                                   const float* __restrict__ twc,
                                                   const float* __restrict__ tws,
                                                   float2* __restrict__ spec) {
  const int wtile   = blockIdx.x * 8 + (threadIdx.x >> 5);  // 16*65 = 1040
  const int rowbase = (wtile / 65) * 64;
  const int f0      = (wtile % 65) * 16;
  const int l       = threadIdx.x & 31;
  const int col     = l & 15;
  const int khalf   = (l >> 4) << 1;                 // f32-WMMA K split
  v8f acc_c[4] = {}; v8f acc_s[4] = {};
  for (int t0 = 0; t0 < NT; t0 += 4) {
    const int tA = t0 + khalf;
    v2f bc, bs;
    bc.x = twc[(size_t)tA * NFP2 + f0 + col];
    bc.y = twc[(size_t)(tA + 1) * NFP2 + f0 + col];
    bs.x = tws[(size_t)tA * NFP2 + f0 + col];
    bs.y = tws[(size_t)(tA + 1) * NFP2 + f0 + col];
#pragma unroll
    for (int mt = 0; mt < 4; ++mt) {
      v2f a;
      a.x = At[(size_t)tA * NROW + rowbase + mt * 16 + col];
      a.y = At[(size_t)(tA + 1) * NROW + rowbase + mt * 16 + col];
      acc_c[mt] = __builtin_amdgcn_wmma_f32_16x16x4_f32(false, a, false, bc,
                                                        (short)0, acc_c[mt], false, false);
      acc_s[mt] = __builtin_amdgcn_wmma_f32_16x16x4_f32(false, a, false, bs,
                                                        (short)0, acc_s[mt], false, false);
    }
  }
  const int fcol = f0 + col;
  if (fcol < NF) {                         // re and im live in the same lane
    const int mrow = (l < 16) ? 0 : 8;     // C/D: VGPR r holds rows {r, r+8}
#pragma unroll
    for (int mt = 0; mt < 4; ++mt)
#pragma unroll
      for (int r = 0; r < 8; ++r) {
        const int row = rowbase + mt * 16 + r + mrow;
        spec[(size_t)row * NF + fcol] = make_float2(acc_c[mt][r], acc_s[mt][r]);
      }
  }
}

// -------------------------------------------------------------------- K2 ---
// 128-point DFT across detectors for one (b,f); folds propagation mask * fw;
// repacks into the WMMA A-panel a3[d][f][16]: rows 0-7 = Re(b), 8-15 = Im(b).
__global__ __launch_bounds__(128) void k_det_dft(const float2* __restrict__ spec,
                                                 float* __restrict__ a3) {
  __shared__ float2 sh[ND];
  __shared__ float2 tab[ND];
  const int bf = blockIdx.x;
  const int b = bf / NFP, f = bf % NFP;
  const int kd = threadIdx.x;
  if (f >= NF) {                          // K padding: zero-fill, block-uniform
    a3[((size_t)kd * NFP + f) * 16 + b]     = 0.f;
    a3[((size_t)kd * NFP + f) * 16 + 8 + b] = 0.f;
    return;
  }
  {
    const float rev = (float)kd * (1.0f / 128.0f);
    tab[kd] = make_float2(__builtin_amdgcn_cosf(rev), __builtin_amdgcn_sinf(rev));
    sh[kd]  = spec[((size_t)b * ND + kd) * NF + f];
  }
  __syncthreads();
  float re = 0.f, im = 0.f;
  int idx = 0;
  for (int d = 0; d < ND; ++d) {
    const float2 cs = tab[idx];
    const float2 v  = sh[d];
    re += v.x * cs.x + v.y * cs.y;        // * e^{-i 2pi kd d / 128}
    im += v.y * cs.x - v.x * cs.y;
    idx = (idx + kd) & (ND - 1);
  }
  const float w     = (float)f * OMEGA_SCALE * INV_C;
  const float kxv   = (float)((kd < 64) ? kd : kd - 128) * KX_SCALE;
  const float kz_sq = fmaf(w, w, -kxv * kxv);
  const float fw    = (f == 0 || f == NF - 1) ? 1.0f : 2.0f;
  const float mfw   = (kz_sq > 0.f) ? fw : 0.f;
  a3[((size_t)kd * NFP + f) * 16 + b]     = re * mfw;
  a3[((size_t)kd * NFP + f) * 16 + 8 + b] = im * mfw;
}

// -------------------------------------------------------------------- K3 ---
// band[b,y,d] = sum_f sf[b,d,f] * e^{i kz(d,f) y}. One wave per (d, y-tile).
// Stacked-complex: D1 = [Re;Im]*cos, D2 = [Re;Im]*sin -> 2 WMMAs per K-step.
__global__ __launch_bounds__(256) void k_band_wmma(const float* __restrict__ a3,
                                                   float2* __restrict__ band) {
  const int tile  = blockIdx.x * 8 + (threadIdx.x >> 5);   // 2048 tiles
  const int d     = tile >> 4;
  const int ybase = (tile & 15) << 4;
  const int l     = threadIdx.x & 31;
  const int m     = l & 15;                  // A row / B column within tile
  const int khalf = (l >> 4) << 1;
  const float kxv = (float)((d < 64) ? d : d - 128) * KX_SCALE;
  const float kx2 = kxv * kxv;
  const float yv  = 1.0e-3f + (float)(ybase + m) * 1.5e-4f;
  v8f acc_c = {}; v8f acc_s = {};
  for (int k0 = 0; k0 < NFP; k0 += 4) {
    const int fa = k0 + khalf;
    v2f a, bc, bs;
    a.x = a3[((size_t)d * NFP + fa)     * 16 + m];
    a.y = a3[((size_t)d * NFP + fa + 1) * 16 + m];
#pragma unroll
    for (int j = 0; j < 2; ++j) {
      const int   f     = fa + j;
      const float w     = (float)f * (OMEGA_SCALE * INV_C);
      const float kz_sq = fmaf(w, w, -kx2);
      const float kz    = (kz_sq > 0.f) ? __builtin_amdgcn_sqrtf(kz_sq) : 0.f;
      float sv, cv; fast_cis(kz * yv, &sv, &cv);
      const float valid = (f < NF) ? 1.f : 0.f;   // zero padded K rows
      if (j == 0) { bc.x = cv * valid; bs.x = sv * valid; }
      else        { bc.y = cv * valid; bs.y = sv * valid; }
    }
    acc_c = __builtin_amdgcn_wmma_f32_16x16x4_f32(false, a, false, bc,
                                                  (short)0, acc_c, false, false);
    acc_s = __builtin_amdgcn_wmma_f32_16x16x4_f32(false, a, false, bs,
                                                  (short)0, acc_s, false, false);
  }
#pragma unroll
  for (int r = 0; r < 8; ++r) {
    const float c_lo = acc_c[r], s_lo = acc_s[r];
    const float c_hi = __shfl_xor(c_lo, 16, 32);
    const float s_hi = __shfl_xor(s_lo, 16, 32);
    if (l < 16) {                         // b = r, y = ybase + l
      const float re = c_lo - s_hi;       // Re*cos - Im*sin
      const float im = s_lo + c_hi;       // Re*sin + Im*cos
      const int row = r * NY_ + ybase + l;
      band[(size_t)d * (NB * NY_) + row] = make_float2(re, im);
    }
  }
}

// -------------------------------------------------------------------- K4 ---
// img[b,y,x] = | sum_d band[b,y,d] * e^{i kx(d) x} | * inv_norm.
__global__ __launch_bounds__(256) void k_img_wmma(const float* __restrict__ band,
                                                  const float* __restrict__ stats,
                                                  float* __restrict__ out) {
  const int tile  = blockIdx.x * 8 + (threadIdx.x >> 5);   // 4096 tiles
  const int rbase = (tile >> 4) << 3;        // 8 (b,y)-rows per tile
  const int xbase = (tile & 15) << 4;
  const int l     = threadIdx.x & 31;
  const int m     = l & 15;
  const int khalf = (l >> 4) << 1;
  const int row   = rbase + (m & 7);
  const int isIm  = m >> 3;
  const float xv  = (float)(xbase + m) * 1.5e-4f;  // xs = x*dx (x0 == array_x0)
  const float inv_norm = stats[16];
  v8f acc_c = {}; v8f acc_s = {};
  for (int k0 = 0; k0 < ND; k0 += 4) {
    v2f a, bc, bs;
#pragma unroll
    for (int j = 0; j < 2; ++j) {
      const int   dk  = k0 + khalf + j;
      const float av  = band[((size_t)dk * (NB * NY_) + row) * 2 + isIm];
      const float kxv = (float)((dk < 64) ? dk : dk - 128) * KX_SCALE;
      float sv, cv; fast_cis(kxv * xv, &sv, &cv);
      if (j == 0) { a.x = av; bc.x = cv; bs.x = sv; }
      else        { a.y = av; bc.y = cv; bs.y = sv; }
    }
    acc_c = __builtin_amdgcn_wmma_f32_16x16x4_f32(false, a, false, bc,
                                                  (short)0, acc_c, false, false);
    acc_s = __builtin_amdgcn_wmma_f32_16x16x4_f32(false, a, false, bs,
                                                  (short)0, acc_s, false, false);
  }
#pragma unroll
  for (int r = 0; r < 8; ++r) {
    const float c_lo = acc_c[r], s_lo = acc_s[r];
    const float c_hi = __shfl_xor(c_lo, 16, 32);
    const float s_hi = __shfl_xor(s_lo, 16, 32);
    if (l < 16) {
      const float re = c_lo - s_hi;
      const float im = s_lo + c_hi;
      out[(size_t)(rbase + r) * NX_ + xbase + l] =
          sqrtf(re * re + im * im) * inv_norm;
    }
  }
}

// ---------------------------------------------------------------------------
extern "C" void kernel_launch(void* const* d_in, const int* in_sizes, int n_in,
                              void* d_out, int out_size, void* d_ws, size_t ws_size,
                              hipStream_t stream) {
  (void)in_sizes; (void)n_in; (void)out_size; (void)ws_size;
  const float* sino = (const float*)d_in[0];
  float* out = (float*)d_out;

  // Workspace layout (~33.8 MB). twc/tws die after k_time_wmma and are
  // reused (stream-ordered) for a3 and band.
  float*  stats = (float*)d_ws;                                    // 17 f32
  float*  At    = (float*)((char*)d_ws + 256);                     // 2048*1024 f32
  float*  twc   = At + (size_t)NT * NROW;                          // 2048*1040 f32
  float*  tws   = twc + (size_t)NT * NFP2;                         // 2048*1040 f32
  float2* spec  = (float2*)(tws + (size_t)NT * NFP2);              // 1024*1025 c64
  float*  a3    = twc;     // 128*1028*16 f32 (8.42 MB <= 8.52 MB) alias
  float2* band  = (float2*)tws;  // 128*2048 c64 (2.1 MB) alias

  k_stats    <<<NB, 256, 0, stream>>>(sino, stats);
  k_pack     <<<32 * 64, 256, 0, stream>>>(sino, stats, At);
  k_twgen    <<<NT, 256, 0, stream>>>(twc, tws);
  k_time_wmma<<<1040 / 8, 256, 0, stream>>>(At, twc, tws, spec);
  k_det_dft  <<<NB * NFP, 128, 0, stream>>>((const float2*)spec, a3);
  k_band_wmma<<<(ND * 16) / 8, 256, 0, stream>>>(a3, band);
  k_img_wmma <<<(NB * (NY_ / 8) * (NX_ / 16)) / 8, 256, 0, stream>>>(
      (const float*)band, stats, out);
}